// EdgeClassifier_62423054680546
// MI455X (gfx1250) — compile-verified
//
#include <hip/hip_runtime.h>
#include <hip/hip_bf16.h>

// ---------------------------------------------------------------------------
// EdgeClassifier GNN for MI455X (gfx1250, wave32, WMMA)
//   conv1: h1 = relu(GCNagg(x @ W1) + b1)
//   conv2: h2 = GCNagg(h1 @ W2) + b2
//   out   = relu([h2[row]|h2[col]|eAttr] @ mW1 + mb1) @ mW2 + mb2
// All matmuls via V_WMMA_F32_16X16X4_F32 (full f32 precision).
// Weights are staged TRANSPOSED in LDS so each B fragment is one aligned
// ds_load_b64 (no repack movs), with pad strides chosen bank-conflict-free
// for the WMMA lane pattern.
// ---------------------------------------------------------------------------

#define N_NODES 50000
#define N_EDGES 800000
#define HID 64
#define E_FEAT 16

#define PAD_NG 68    // node-gemm LDS row stride (floats): bank step 4, even
#define PAD_E  148   // edge-mlp LDS row stride (floats): bank step 20, even

typedef float v2f __attribute__((ext_vector_type(2)));
typedef float v8f __attribute__((ext_vector_type(8)));

__device__ __forceinline__ v8f wmma_f32x4(v2f a, v2f b, v8f c) {
    // 8 args: (neg_a, A, neg_b, B, c_mod, C, reuse_a, reuse_b)
    return __builtin_amdgcn_wmma_f32_16x16x4_f32(false, a, false, b, (short)0, c,
                                                 false, false);
}

__device__ __forceinline__ v2f lds_b64(const float* p) {
    return *(const v2f*)p;  // even-float offset -> ds_load_b64
}

// ---------------- degree / normalization --------------------------------
__global__ __launch_bounds__(256) void deg_kernel(const int* __restrict__ col,
                                                  int* __restrict__ deg) {
    int e = blockIdx.x * 256 + threadIdx.x;
    if (e < N_EDGES) atomicAdd(&deg[col[e]], 1);
}

__global__ __launch_bounds__(256) void dinv_kernel(const int* __restrict__ deg,
                                                   float* __restrict__ dinv) {
    int i = blockIdx.x * 256 + threadIdx.x;
    if (i < N_NODES) dinv[i] = rsqrtf((float)(deg[i] + 1));  // +1 self loop
}

// ---------------- node GEMM: out[M,64] = A[M,64] @ W[64,64] -------------
// 8 waves/block; each wave computes a 16-row tile over all 64 columns.
__global__ __launch_bounds__(256) void node_gemm_kernel(
    const float* __restrict__ A, const float* __restrict__ W,
    float* __restrict__ out, int M) {
    __shared__ float sW[64 * PAD_NG];  // transposed: sW[n*PAD + k]
    int tid = threadIdx.x;
#pragma unroll
    for (int i = 0; i < 16; ++i) {
        int idx = tid + 256 * i;  // 4096 elements, coalesced global read
        sW[(idx & 63) * PAD_NG + (idx >> 6)] = W[idx];
    }
    __syncthreads();

    int wave = tid >> 5, lane = tid & 31;
    int tile = blockIdx.x * 8 + wave;
    if (tile * 16 >= M) return;
    int n  = lane & 15;                 // A: row m == n slot; B/C: column n
    int kh = (lane >> 4) << 1;          // K sub-pair: 0 or 2 (ISA 16x4 layout)
    const float* arow = A + (tile * 16 + n) * 64;

    v8f c[4];
#pragma unroll
    for (int nt = 0; nt < 4; ++nt)
#pragma unroll
        for (int r = 0; r < 8; ++r) c[nt][r] = 0.0f;

#pragma unroll
    for (int ks = 0; ks < 16; ++ks) {
        int f0 = 4 * ks + kh;
        v2f a = *(const v2f*)(arow + f0);       // global_load_b64
#pragma unroll
        for (int nt = 0; nt < 4; ++nt) {
            v2f b = lds_b64(&sW[(nt * 16 + n) * PAD_NG + f0]);
            c[nt] = wmma_f32x4(a, b, c[nt]);
        }
    }
    int mbase = tile * 16 + ((lane >> 4) << 3);  // rows 0..7 / 8..15
#pragma unroll
    for (int nt = 0; nt < 4; ++nt)
#pragma unroll
        for (int r = 0; r < 8; ++r)
            out[(mbase + r) * 64 + nt * 16 + n] = c[nt][r];
}

// ---------------- edge scatter: hacc[col] += t[row] * dinv[r]*dinv[c] ----
__global__ __launch_bounds__(256) void scatter_kernel(
    const float* __restrict__ t, const int* __restrict__ row,
    const int* __restrict__ col, const float* __restrict__ dinv,
    float* __restrict__ hacc) {
    int gid = blockIdx.x * 256 + threadIdx.x;  // E*16 threads, 4 feats each
    int e = gid >> 4, q = (gid & 15) << 2;
    int r = row[e], c = col[e];
    float nrm = dinv[r] * dinv[c];
    const float* src = t + r * 64 + q;
    float* dst = hacc + c * 64 + q;
#pragma unroll
    for (int j = 0; j < 4; ++j) atomicAdd(&dst[j], src[j] * nrm);
}

// ---------------- finish: h = act(hacc + t*dinv^2 + b) -------------------
__global__ __launch_bounds__(256) void finish_kernel(
    const float* __restrict__ hacc, const float* __restrict__ t,
    const float* __restrict__ dinv, const float* __restrict__ bias,
    float* __restrict__ h, int relu) {
    int gid = blockIdx.x * 256 + threadIdx.x;  // N*64 threads
    int i = gid >> 6, k = gid & 63;
    float di = dinv[i];
    float v = hacc[gid] + t[gid] * di * di + bias[k];
    h[gid] = (relu && v < 0.0f) ? 0.0f : v;
}

// ---------------- edge MLP ----------------------------------------------
// 8 waves/block, 16 edges/wave. A = [h2[row] | h2[col] | eAttr] (144 feats)
// gathered straight into WMMA A fragments; mW1 staged transposed in LDS.
__global__ __launch_bounds__(256) void edge_mlp_kernel(
    const float* __restrict__ h2, const int* __restrict__ erow,
    const int* __restrict__ ecol, const float* __restrict__ eattr,
    const float* __restrict__ mW1, const float* __restrict__ mb1,
    const float* __restrict__ mW2, const float* __restrict__ mb2,
    float* __restrict__ out) {
    __shared__ float sW[64 * PAD_E];  // transposed: sW[n*PAD_E + k], ~37.9 KB
    int tid = threadIdx.x;
#pragma unroll
    for (int i = 0; i < 36; ++i) {
        int idx = tid + 256 * i;  // 9216 elements, coalesced global read
        sW[(idx & 63) * PAD_E + (idx >> 6)] = mW1[idx];
    }
    __syncthreads();

    int wave = tid >> 5, lane = tid & 31;
    int tile = blockIdx.x * 8 + wave;      // 50000 tiles exactly
    int n  = lane & 15;
    int kh = (lane >> 4) << 1;
    int e = tile * 16 + n;
    const float* hr = h2 + erow[e] * 64;
    const float* hc = h2 + ecol[e] * 64;
    const float* ea = eattr + e * 16;

    v8f c[4];
#pragma unroll
    for (int nt = 0; nt < 4; ++nt) {
        float bv = mb1[nt * 16 + n];       // bias folded into C init
#pragma unroll
        for (int r = 0; r < 8; ++r) c[nt][r] = bv;
    }

    // K-segment 1: features 0..63 <- h2[row]
#pragma unroll
    for (int ks = 0; ks < 16; ++ks) {
        int f0 = 4 * ks + kh;
        v2f a = *(const v2f*)(hr + f0);
#pragma unroll
        for (int nt = 0; nt < 4; ++nt) {
            v2f b = lds_b64(&sW[(nt * 16 + n) * PAD_E + f0]);
            c[nt] = wmma_f32x4(a, b, c[nt]);
        }
    }
    // K-segment 2: features 64..127 <- h2[col]
#pragma unroll
    for (int ks = 0; ks < 16; ++ks) {
        int f0 = 4 * ks + kh;
        v2f a = *(const v2f*)(hc + f0);
#pragma unroll
        for (int nt = 0; nt < 4; ++nt) {
            v2f b = lds_b64(&sW[(nt * 16 + n) * PAD_E + 64 + f0]);
            c[nt] = wmma_f32x4(a, b, c[nt]);
        }
    }
    // K-segment 3: features 128..143 <- eAttr
#pragma unroll
    for (int ks = 0; ks < 4; ++ks) {
        int f0 = 4 * ks + kh;
        v2f a = *(const v2f*)(ea + f0);
#pragma unroll
        for (int nt = 0; nt < 4; ++nt) {
            v2f b = lds_b64(&sW[(nt * 16 + n) * PAD_E + 128 + f0]);
            c[nt] = wmma_f32x4(a, b, c[nt]);
        }
    }

    // layer 2: out[e] = relu(z[e,:]) . mW2 + mb2  (reduce across the 16 lanes
    // of each half-wave; rows live in the 8 accumulator registers)
    float acc[8];
#pragma unroll
    for (int r = 0; r < 8; ++r) acc[r] = 0.0f;
#pragma unroll
    for (int nt = 0; nt < 4; ++nt) {
        float w = mW2[nt * 16 + n];
#pragma unroll
        for (int r = 0; r < 8; ++r) {
            float z = c[nt][r];
            z = z > 0.0f ? z : 0.0f;
            acc[r] += z * w;
        }
    }
#pragma unroll
    for (int mask = 1; mask <= 8; mask <<= 1)
#pragma unroll
        for (int r = 0; r < 8; ++r) acc[r] += __shfl_xor(acc[r], mask, 32);

    if (n == 0) {  // lane 0 -> rows 0..7, lane 16 -> rows 8..15
        float b2v = mb2[0];
        int base = tile * 16 + ((lane >> 4) << 3);
#pragma unroll
        for (int r = 0; r < 8; ++r) out[base + r] = acc[r] + b2v;
    }
}

// ---------------------------------------------------------------------------
extern "C" void kernel_launch(void* const* d_in, const int* in_sizes, int n_in,
                              void* d_out, int out_size, void* d_ws, size_t ws_size,
                              hipStream_t stream) {
    (void)in_sizes; (void)n_in; (void)out_size; (void)ws_size;
    const float* x     = (const float*)d_in[0];
    const int*   eIdx  = (const int*)d_in[1];
    const int*   erow  = eIdx;
    const int*   ecol  = eIdx + N_EDGES;
    const float* eattr = (const float*)d_in[2];
    const float* W1    = (const float*)d_in[3];
    const float* b1    = (const float*)d_in[4];
    const float* W2    = (const float*)d_in[5];
    const float* b2    = (const float*)d_in[6];
    const float* mW1   = (const float*)d_in[7];
    const float* mb1   = (const float*)d_in[8];
    const float* mW2   = (const float*)d_in[9];
    const float* mb2   = (const float*)d_in[10];
    float* out = (float*)d_out;

    // workspace layout (~38.8 MB)
    float* t    = (float*)d_ws;               // N*64  : pre-aggregation x@W
    float* hacc = t + (size_t)N_NODES * 64;   // N*64  : scatter accumulator
    float* h    = hacc + (size_t)N_NODES * 64;// N*64  : node features
    float* dinv = h + (size_t)N_NODES * 64;   // N
    int*   deg  = (int*)(dinv + N_NODES);     // N

    const int GEMM_BLOCKS = (N_NODES / 16 + 7) / 8;        // 391
    const int SCAT_BLOCKS = (N_EDGES * 16) / 256;          // 50000
    const int FIN_BLOCKS  = (N_NODES * 64) / 256;          // 12500

    hipMemsetAsync(deg, 0, N_NODES * sizeof(int), stream);
    hipMemsetAsync(hacc, 0, (size_t)N_NODES * 64 * sizeof(float), stream);

    deg_kernel<<<(N_EDGES + 255) / 256, 256, 0, stream>>>(ecol, deg);
    dinv_kernel<<<(N_NODES + 255) / 256, 256, 0, stream>>>(deg, dinv);

    // conv1
    node_gemm_kernel<<<GEMM_BLOCKS, 256, 0, stream>>>(x, W1, t, N_NODES);
    scatter_kernel<<<SCAT_BLOCKS, 256, 0, stream>>>(t, erow, ecol, dinv, hacc);
    finish_kernel<<<FIN_BLOCKS, 256, 0, stream>>>(hacc, t, dinv, b1, h, 1);

    // conv2
    hipMemsetAsync(hacc, 0, (size_t)N_NODES * 64 * sizeof(float), stream);
    node_gemm_kernel<<<GEMM_BLOCKS, 256, 0, stream>>>(h, W2, t, N_NODES);
    scatter_kernel<<<SCAT_BLOCKS, 256, 0, stream>>>(t, erow, ecol, dinv, hacc);
    finish_kernel<<<FIN_BLOCKS, 256, 0, stream>>>(hacc, t, dinv, b2, h, 0);

    // edge MLP
    edge_mlp_kernel<<<N_EDGES / 128, 256, 0, stream>>>(h, erow, ecol, eattr,
                                                       mW1, mb1, mW2, mb2, out);
}